// ConditionalSplineSQ2D_57432302682418
// MI455X (gfx1250) — compile-verified
//
#include <hip/hip_runtime.h>

typedef float v2f __attribute__((ext_vector_type(2)));
typedef float v4f __attribute__((ext_vector_type(4)));
typedef float v8f __attribute__((ext_vector_type(8)));

#define B_DIM 4096
#define G_DIM 31
#define P_DIM 8
#define C_DIM 36

__global__ void zero_kernel(float* __restrict__ out, int n) {
    int i = blockIdx.x * blockDim.x + threadIdx.x;
    if (i < n) out[i] = 0.0f;
}

// Branch-free S[i][k]: c(i,k) = i*(17-i)/2 + (k-i) lies in [0,35] for ALL
// i,k in [0,7] (min 0 at i=0,k=0; max 35 at i=7,k=7), so always load and
// mask the lower triangle to zero with a select -- no exec divergence.
__device__ __forceinline__ float sval_masked(const float* __restrict__ cgh, int i, int k) {
    int c = ((i * (17 - i)) >> 1) + (k - i);
    float v = cgh[c];
    return (i <= k) ? v : 0.0f;
}

__global__ void __launch_bounds__(256)
quadform_kernel(const float* __restrict__ param,
                const float* __restrict__ coeffs,
                float* __restrict__ out) {
    // Pre-swizzled A-matrix lane values for every h of this block's g:
    // identical for all 8 waves, so build once in LDS. 31*32*4 f32 = 15.9 KB.
    __shared__ float sA[G_DIM][32][4];

    const int lane = threadIdx.x & 31;
    const int wave = threadIdx.x >> 5;
    const int m    = lane & 15;   // tile row / column index
    const int hi   = lane >> 4;   // half-wave: K pair {0,1} vs {2,3}
    const int koff = hi << 1;

    const int g    = blockIdx.y;
    const int base = blockIdx.x * 256 + wave * 32;  // 32 batch rows per wave
    const int b0   = base + m;
    const int b1   = b0 + 16;

    // ---- one-time LDS build of WMMA A-operand (S, rows replicated) ----
    for (int idx = threadIdx.x; idx < G_DIM * 32; idx += 256) {
        const int h  = idx >> 5;
        const int l  = idx & 31;
        const int lk = (l >> 4) << 1;     // that lane's K pair base
        const int ls = l & 7;             // replicated S row
        const float* cgh = coeffs + ((size_t)g * G_DIM + h) * C_DIM;
        sA[h][l][0] = sval_masked(cgh, ls, lk + 0);
        sA[h][l][1] = sval_masked(cgh, ls, lk + 1);
        sA[h][l][2] = sval_masked(cgh, ls, lk + 4);
        sA[h][l][3] = sval_masked(cgh, ls, lk + 5);
    }
    __syncthreads();

    float acc0 = 0.0f, acc1 = 0.0f;

    for (int h = 0; h < G_DIM; ++h) {
        // A-operand: one ds_load_b128 per lane, split into the two K-halves.
        v4f a = *(const v4f*)(&sA[h][lane][0]);
        v2f aLo; aLo.x = a.x; aLo.y = a.y;
        v2f aHi; aHi.x = a.z; aHi.y = a.w;

        // B-operand (4x16 fp32): column n = m is batch row b; lane supplies
        // k = koff + v. Contiguous float2 loads -> each param element read once.
        const float* p0 = param + (((size_t)b0 * G_DIM + g) * G_DIM + h) * (size_t)P_DIM;
        const float* p1 = param + (((size_t)b1 * G_DIM + g) * G_DIM + h) * (size_t)P_DIM;
        v2f bA = *(const v2f*)(p0 + koff);
        v2f bB = *(const v2f*)(p0 + koff + 4);
        v2f bC = *(const v2f*)(p1 + koff);
        v2f bD = *(const v2f*)(p1 + koff + 4);

        __builtin_prefetch(p0 + 2 * P_DIM, 0, 3);
        __builtin_prefetch(p1 + 2 * P_DIM, 0, 3);

        // Y = S * P^T, K=8 via two chained 16x16x4 fp32 WMMAs per 16-b tile.
        // Interleaved so consecutive WMMAs are independent (no RAW chain).
        v8f d0 = {};
        v8f d1 = {};
        d0 = __builtin_amdgcn_wmma_f32_16x16x4_f32(false, aLo, false, bA, (short)0, d0, false, false);
        d1 = __builtin_amdgcn_wmma_f32_16x16x4_f32(false, aLo, false, bC, (short)0, d1, false, false);
        d0 = __builtin_amdgcn_wmma_f32_16x16x4_f32(false, aHi, false, bB, (short)0, d0, false, false);
        d1 = __builtin_amdgcn_wmma_f32_16x16x4_f32(false, aHi, false, bD, (short)0, d1, false, false);

        // Closing dot p . (S p): lane owns k in {koff,koff+1,koff+4,koff+5};
        // Y[r,b] sits in D VGPR r for all lanes (replicated-S trick).
        float y0 = hi ? d0[2] : d0[0];
        float y1 = hi ? d0[3] : d0[1];
        float y4 = hi ? d0[6] : d0[4];
        float y5 = hi ? d0[7] : d0[5];
        acc0 += y0 * bA.x + y1 * bA.y + y4 * bB.x + y5 * bB.y;

        float z0 = hi ? d1[2] : d1[0];
        float z1 = hi ? d1[3] : d1[1];
        float z4 = hi ? d1[6] : d1[4];
        float z5 = hi ? d1[7] : d1[5];
        acc1 += z0 * bC.x + z1 * bC.y + z4 * bD.x + z5 * bD.y;
    }

    // lanes n and n^16 hold partials for the same b: merge half-waves.
    float o0 = __shfl(acc0, lane ^ 16, 32);
    float o1 = __shfl(acc1, lane ^ 16, 32);
    float s0 = acc0 + o0;
    float s1 = acc1 + o1;
    if (lane < 16) atomicAdd(&out[base + lane], s0);  // b = base + lane
    else           atomicAdd(&out[base + lane], s1);  // b = base+16 + (lane-16)
}

extern "C" void kernel_launch(void* const* d_in, const int* in_sizes, int n_in,
                              void* d_out, int out_size, void* d_ws, size_t ws_size,
                              hipStream_t stream) {
    const float* param  = (const float*)d_in[0];   // [B, G, G, P] fp32
    const float* coeffs = (const float*)d_in[1];   // [G, G, C] fp32
    float* out = (float*)d_out;                    // [B] fp32

    zero_kernel<<<(out_size + 255) / 256, 256, 0, stream>>>(out, out_size);

    dim3 grid(B_DIM / 256, G_DIM);   // 16 b-chunks x 31 g-slices = 496 blocks
    quadform_kernel<<<grid, 256, 0, stream>>>(param, coeffs, out);
}